// GCNClassifier_13675175870525
// MI455X (gfx1250) — compile-verified
//
#include <hip/hip_runtime.h>
#include <math.h>

typedef float v2f __attribute__((ext_vector_type(2)));
typedef float v8f __attribute__((ext_vector_type(8)));

#define NNODES 50000
#define NEDGES 500000
#define DIN    128
#define HDIM   256
#define NPAIRS 100000

// ---------------------------------------------------------------- degree prep
__global__ void kzero(float* __restrict__ p, int n) {
  int i = blockIdx.x * blockDim.x + threadIdx.x;
  if (i < n) p[i] = 0.0f;
}

__global__ void kdeg(const int* __restrict__ dst, int e, float* __restrict__ deg) {
  int i = blockIdx.x * blockDim.x + threadIdx.x;
  if (i < e) atomicAdd(&deg[dst[i]], 1.0f);
}

__global__ void kdinv(float* __restrict__ deg, int n) {
  int i = blockIdx.x * blockDim.x + threadIdx.x;
  if (i < n) deg[i] = rsqrtf(deg[i] + 1.0f);   // +1 self-loop; always > 0
}

// ---------------------------------------------------------------- WMMA GEMM
// C[M,Nc] = A[M,K] @ B[K,Nc].  M%16==0, K%4==0, Nc%64==0.
// One wave computes a 16x64 output slab (4 accumulators, A-frag reuse).
__global__ void gemm_wmma(const float* __restrict__ A, const float* __restrict__ B,
                          float* __restrict__ C, int M, int K, int Nc) {
  const int lane = threadIdx.x & 31;
  const int wid  = blockIdx.x * (blockDim.x >> 5) + (threadIdx.x >> 5);
  const int nt4  = Nc >> 6;
  if (wid >= (M >> 4) * nt4) return;            // wave-uniform guard (EXEC stays full)
  const int tm   = wid / nt4;
  const int tn   = (wid % nt4) << 6;
  const int half = lane >> 4;
  const int l15  = lane & 15;

  const float* Ap = A + (long)((tm << 4) + l15) * K;   // A: row = lane&15, k = k+2*half
  v8f c0 = {}; v8f c1 = {}; v8f c2 = {}; v8f c3 = {};

  for (int k = 0; k < K; k += 4) {
    const int kk = k + (half << 1);
    v2f a = *(const v2f*)(Ap + kk);             // A[row][kk], A[row][kk+1]
    const float* B0 = B + (long)kk * Nc + tn + l15;
    const float* B1 = B0 + Nc;
    v2f b;
    b.x = B0[0];  b.y = B1[0];
    c0 = __builtin_amdgcn_wmma_f32_16x16x4_f32(false, a, false, b, (short)0, c0, false, false);
    b.x = B0[16]; b.y = B1[16];
    c1 = __builtin_amdgcn_wmma_f32_16x16x4_f32(false, a, false, b, (short)0, c1, false, false);
    b.x = B0[32]; b.y = B1[32];
    c2 = __builtin_amdgcn_wmma_f32_16x16x4_f32(false, a, false, b, (short)0, c2, false, false);
    b.x = B0[48]; b.y = B1[48];
    c3 = __builtin_amdgcn_wmma_f32_16x16x4_f32(false, a, false, b, (short)0, c3, false, false);
  }
  // C/D layout: VGPR i -> row = i + 8*half, col = lane&15
  float* Cp = C + (long)((tm << 4) + (half << 3)) * Nc + tn + l15;
  #pragma unroll
  for (int i = 0; i < 8; ++i) {
    Cp[(long)i * Nc + 0]  = c0[i];
    Cp[(long)i * Nc + 16] = c1[i];
    Cp[(long)i * Nc + 32] = c2[i];
    Cp[(long)i * Nc + 48] = c3[i];
  }
}

// ---------------------------------------------------------------- aggregation
// self-loop contribution initializes the output (no atomics needed)
__global__ void kinit_agg(const float* __restrict__ h, const float* __restrict__ dinv,
                          float* __restrict__ out, int n) {
  int i = blockIdx.x * blockDim.x + threadIdx.x;
  if (i < n) {
    float d = dinv[i >> 8];                     // HDIM == 256
    out[i] = h[i] * d * d;
  }
}

// 64 threads per edge, float4 gather + 4 float atomics (L2-resident: h fits in 192MB L2)
__global__ void kedge_agg(const float* __restrict__ h, const int* __restrict__ src,
                          const int* __restrict__ dst, const float* __restrict__ dinv,
                          float* __restrict__ out, int e) {
  int tid = blockIdx.x * blockDim.x + threadIdx.x;
  int eid = tid >> 6;
  if (eid >= e) return;
  int c4 = (tid & 63) << 2;
  int s = src[eid], d = dst[eid];
  float w = dinv[s] * dinv[d];
  float4 hv = *(const float4*)(h + (long)s * HDIM + c4);
  float* op = out + (long)d * HDIM + c4;
  atomicAdd(op + 0, hv.x * w);
  atomicAdd(op + 1, hv.y * w);
  atomicAdd(op + 2, hv.z * w);
  atomicAdd(op + 3, hv.w * w);
}

__global__ void kbias_relu(float* __restrict__ buf, const float* __restrict__ b, int n) {
  int i = blockIdx.x * blockDim.x + threadIdx.x;
  if (i < n) {
    float x = buf[i] + b[i & (HDIM - 1)];
    buf[i] = x > 0.0f ? x : 0.0f;
  }
}

// ---------------------------------------------------------------- fused MLP head
// Per wave: 16 pairs. A row = [xl[y0] | xr[y1]] (K=512, gathered in the frag loader),
// 16 accumulators = all 256 hidden cols, then relu + dot(fc2_W) in registers,
// shfl_xor reduce across the 16-lane half, sigmoid, store.
__global__ void kfc_fused(const float* __restrict__ xl, const float* __restrict__ xr,
                          const int* __restrict__ y,
                          const float* __restrict__ fcW, const float* __restrict__ fcb,
                          const float* __restrict__ w2, const float* __restrict__ b2,
                          float* __restrict__ out) {
  const int lane = threadIdx.x & 31;
  const int wid  = blockIdx.x * (blockDim.x >> 5) + (threadIdx.x >> 5);
  if (wid >= NPAIRS / 16) return;               // wave-uniform
  const int half = lane >> 4;
  const int l15  = lane & 15;
  const int p0   = wid << 4;
  const int pr   = p0 + l15;                    // this lane's A row (pair index)
  const int y0   = y[pr * 2 + 0];
  const int y1   = y[pr * 2 + 1];
  const float* AL = xl + (long)y0 * HDIM;
  const float* AR = xr + (long)y1 * HDIM;

  v8f c[16];
  #pragma unroll
  for (int j = 0; j < 16; ++j) c[j] = (v8f){};

  for (int k = 0; k < 2 * HDIM; k += 4) {
    const int kk = k + (half << 1);             // never straddles the 256 boundary
    v2f a = (kk < HDIM) ? *(const v2f*)(AL + kk) : *(const v2f*)(AR + kk - HDIM);
    const float* B0 = fcW + (long)kk * HDIM + l15;
    const float* B1 = B0 + HDIM;
    #pragma unroll
    for (int j = 0; j < 16; ++j) {
      v2f b;
      b.x = B0[j * 16];
      b.y = B1[j * 16];
      c[j] = __builtin_amdgcn_wmma_f32_16x16x4_f32(false, a, false, b, (short)0, c[j], false, false);
    }
  }

  float part[8];
  #pragma unroll
  for (int i = 0; i < 8; ++i) part[i] = 0.0f;
  #pragma unroll
  for (int j = 0; j < 16; ++j) {
    int col = (j << 4) + l15;
    float bb = fcb[col];
    float ww = w2[col];                         // fc2_W is [256,1]
    #pragma unroll
    for (int i = 0; i < 8; ++i) {
      float u = c[j][i] + bb;
      u = u > 0.0f ? u : 0.0f;
      part[i] += u * ww;
    }
  }
  // reduce over the 16 lanes of each half (rows differ between halves)
  #pragma unroll
  for (int i = 0; i < 8; ++i) {
    #pragma unroll
    for (int off = 1; off < 16; off <<= 1)
      part[i] += __shfl_xor(part[i], off, 32);
  }
  if (l15 == 0) {
    float bias2 = b2[0];
    int r0 = p0 + (half << 3);                  // half 0 -> rows 0..7, half 1 -> rows 8..15
    #pragma unroll
    for (int i = 0; i < 8; ++i) {
      float lg = part[i] + bias2;
      out[r0 + i] = 1.0f / (1.0f + expf(-lg));
    }
  }
}

// ---------------------------------------------------------------- launch
extern "C" void kernel_launch(void* const* d_in, const int* in_sizes, int n_in,
                              void* d_out, int out_size, void* d_ws, size_t ws_size,
                              hipStream_t stream) {
  const float* x_s  = (const float*)d_in[0];
  const float* x_t  = (const float*)d_in[1];
  const int*   es   = (const int*)d_in[2];     // [2,E]: src = es, dst = es+E
  const int*   et   = (const int*)d_in[3];
  const int*   y    = (const int*)d_in[4];     // [P,2]
  const float* W1   = (const float*)d_in[5];
  const float* b1   = (const float*)d_in[6];
  const float* W2   = (const float*)d_in[7];
  const float* b2g  = (const float*)d_in[8];
  const float* fcW  = (const float*)d_in[9];
  const float* fcb  = (const float*)d_in[10];
  const float* fc2W = (const float*)d_in[11];
  const float* fc2b = (const float*)d_in[12];
  float* out = (float*)d_out;

  float* dinv_s = (float*)d_ws;
  float* dinv_t = dinv_s + NNODES;
  float* bufA   = dinv_t + NNODES;                 // GEMM output (reused every layer)
  float* bufB   = bufA + (long)NNODES * HDIM;      // layer-1 output / xr final
  float* bufC   = bufB + (long)NNODES * HDIM;      // xl final

  const int TPB = 256;
  const int gN  = (NNODES + TPB - 1) / TPB;
  const int gE  = (NEDGES + TPB - 1) / TPB;
  const int nNH = NNODES * HDIM;
  const int gNH = (nNH + TPB - 1) / TPB;
  const int gEA = (int)(((long)NEDGES * 64) / TPB);            // exact: 125000
  const int gGW = ((NNODES / 16) * (HDIM / 64) + 7) / 8;       // 8 waves/block

  // degrees -> dinv for both graphs
  kzero<<<gN, TPB, 0, stream>>>(dinv_s, NNODES);
  kzero<<<gN, TPB, 0, stream>>>(dinv_t, NNODES);
  kdeg<<<gE, TPB, 0, stream>>>(es + NEDGES, NEDGES, dinv_s);
  kdeg<<<gE, TPB, 0, stream>>>(et + NEDGES, NEDGES, dinv_t);
  kdinv<<<gN, TPB, 0, stream>>>(dinv_s, NNODES);
  kdinv<<<gN, TPB, 0, stream>>>(dinv_t, NNODES);

  // one GCN layer: tmp = x@W ; out = scatter(norm * tmp) ; out = relu(out + bias)
  auto gcn = [&](const float* x, int K, const float* W, const float* bias,
                 const int* ei, const float* dinv, float* tmp, float* o) {
    gemm_wmma<<<gGW, TPB, 0, stream>>>(x, W, tmp, NNODES, K, HDIM);
    kinit_agg<<<gNH, TPB, 0, stream>>>(tmp, dinv, o, nNH);
    kedge_agg<<<gEA, TPB, 0, stream>>>(tmp, ei, ei + NEDGES, dinv, o, NEDGES);
    kbias_relu<<<gNH, TPB, 0, stream>>>(o, bias, nNH);
  };

  // source tower -> xl in bufC
  gcn(x_s,  DIN,  W1, b1, es, dinv_s, bufA, bufB);
  gcn(bufB, HDIM, W2, b2g, es, dinv_s, bufA, bufC);
  // target tower -> xr in bufB (bufB is dead after its GEMM is consumed)
  gcn(x_t,  DIN,  W1, b1, et, dinv_t, bufA, bufB);
  gcn(bufB, HDIM, W2, b2g, et, dinv_t, bufA, bufB);

  // fused fc -> relu -> fc2 -> sigmoid
  const int gFC = ((NPAIRS / 16) + 7) / 8;
  kfc_fused<<<gFC, TPB, 0, stream>>>(bufC, bufB, y, fcW, fcb, fc2W, fc2b, out);
}